// EdgeUpdateNetwork_13666585936608
// MI455X (gfx1250) — compile-verified
//
#include <hip/hip_runtime.h>

typedef __attribute__((ext_vector_type(2))) float v2f;
typedef __attribute__((ext_vector_type(8))) float v8f;

#define EPS 1e-5f
#define NEG_SLOPE 0.01f
#define NN 256
#define FF 128
#define CIN 258   // 2*F + 2

// ---------------------------------------------------------------------------
// Kernel 1: fold BN1/BN2/pw/pb into small constant tables in workspace P.
//   P[  0..127] we0_s = w1[:,0]*s1      P[128..255] we1_s = w1[:,1]*s1
//   P[256..383] W0[f] = sum_g pw[0,g]*s2[g]*w2[g,f]
//   P[384..511] W1[f] = sum_g pw[1,g]*s2[g]*w2[g,f]
//   P[512..513] c[o]  = pb[o] + sum_g pw[o,g]*t2[g]
//   P[514..641] s1    P[642..769] t1
// ---------------------------------------------------------------------------
__global__ void __launch_bounds__(128) k_prep(
    const float* __restrict__ w1,
    const float* __restrict__ g1, const float* __restrict__ b1,
    const float* __restrict__ m1, const float* __restrict__ v1,
    const float* __restrict__ w2,
    const float* __restrict__ g2, const float* __restrict__ b2,
    const float* __restrict__ m2, const float* __restrict__ v2,
    const float* __restrict__ pw, const float* __restrict__ pb,
    float* __restrict__ P)
{
    __shared__ float s2sh[FF];
    __shared__ float t2sh[FF];
    const int f = threadIdx.x;           // 0..127
    const float s2 = g2[f] * rsqrtf(v2[f] + EPS);
    s2sh[f] = s2;
    t2sh[f] = b2[f] - m2[f] * s2;
    const float s1 = g1[f] * rsqrtf(v1[f] + EPS);
    const float t1 = b1[f] - m1[f] * s1;
    P[514 + f] = s1;
    P[642 + f] = t1;
    P[f]       = w1[f * CIN + 0] * s1;   // we0 pre-scaled by s1
    P[128 + f] = w1[f * CIN + 1] * s1;   // we1 pre-scaled by s1
    __syncthreads();
    float W0 = 0.f, W1 = 0.f;
    for (int g = 0; g < FF; ++g) {
        const float wg = w2[g * FF + f];
        W0 = fmaf(pw[g]        * s2sh[g], wg, W0);
        W1 = fmaf(pw[FF + g]   * s2sh[g], wg, W1);
    }
    P[256 + f] = W0;
    P[384 + f] = W1;
    if (f < 2) {
        float c = pb[f];
        for (int g = 0; g < FF; ++g) c = fmaf(pw[f * FF + g], t2sh[g], c);
        P[512 + f] = c;
    }
}

// ---------------------------------------------------------------------------
// Kernel 2: hl_s / hr_s via f32 WMMA.
//   hl[m,g] = sum_f node[m,f]*w_l[g,f]  (m = b*N+i, 1024x128 @ 128x128)
// One wave per 16x16 tile; K in steps of 4 with v_wmma_f32_16x16x4_f32.
//   A frag (16x4 f32): lanes 0-15 -> M=lane, K={k,k+1}; lanes 16-31 -> K={k+2,k+3}
//   B frag (4x16 f32): mirrored lane striping, B[k',n] = w_{l|r}[g0+n, k+k']
// After the K loop: hl_s = acc*s1[g]; hr_s = acc*s1[g] + t1[g].
// ---------------------------------------------------------------------------
__global__ void __launch_bounds__(128) k_gemm_hlr(
    const float* __restrict__ node,  // (1024,128)
    const float* __restrict__ w1,    // (128,258)
    const float* __restrict__ P,
    float* __restrict__ HL,          // (1024,128)
    float* __restrict__ HR)          // (1024,128)
{
    const int lane = threadIdx.x & 31;
    const int wid  = blockIdx.x * (blockDim.x >> 5) + (threadIdx.x >> 5); // 0..1023
    const int mat  = wid >> 9;        // 0 = hl (w_l), 1 = hr (w_r)
    const int t    = wid & 511;
    const int m0   = (t >> 3) << 4;   // 64 M tiles
    const int g0   = (t & 7) << 4;    // 8  N tiles
    const int lm   = lane & 15;
    const int koff = (lane >> 4) << 1;

    const float* __restrict__ Arow = node + (m0 + lm) * FF + koff;
    const float* __restrict__ Brow = w1 + (g0 + lm) * CIN + (mat ? 130 : 2) + koff;

    v8f acc = {0.f, 0.f, 0.f, 0.f, 0.f, 0.f, 0.f, 0.f};
#pragma unroll 8
    for (int k = 0; k < FF; k += 4) {
        v2f a = *(const v2f*)(Arow + k);   // 8B aligned: even element offsets
        v2f b = *(const v2f*)(Brow + k);
        acc = __builtin_amdgcn_wmma_f32_16x16x4_f32(
            /*neg_a=*/false, a, /*neg_b=*/false, b,
            /*c_mod=*/(short)0, acc, /*reuse_a=*/false, /*reuse_b=*/false);
    }

    const float s1 = P[514 + g0 + lm];
    const float t1 = mat ? P[642 + g0 + lm] : 0.0f;
    float* __restrict__ D = mat ? HR : HL;
    const int half8 = (lane >> 4) << 3; // lanes 16-31 hold rows M=m0+8..15
#pragma unroll
    for (int r = 0; r < 8; ++r) {
        D[(m0 + half8 + r) * FF + g0 + lm] = fmaf(acc[r], s1, t1);
    }
}

// ---------------------------------------------------------------------------
// Kernel 3: fused edge update. One block per (b, 16x16 (i,j) tile); one
// thread per edge. Tiles of hl_s/hr_s staged in LDS (rows padded to 132
// floats -> conflict-free b128 reads), params staged once. Inner loop:
//   z = leaky(e0*we0_s + e1*we1_s + hl_s + hr_s);  acc_o += W[o,f]*z
// ---------------------------------------------------------------------------
#define ROWP 132   // 128 + 4 pad floats (row stride 528B, 16B aligned)

__global__ void __launch_bounds__(256) k_edge_fused(
    const float* __restrict__ edge,  // (4,256,256,2)
    const float* __restrict__ P,
    const float* __restrict__ HL,
    const float* __restrict__ HR,
    float* __restrict__ out)         // (4,256,256,2)
{
    __shared__ alignas(16) float sHL[16 * ROWP];
    __shared__ alignas(16) float sHR[16 * ROWP];
    __shared__ alignas(16) float sPar[512];

    const int tid = threadIdx.x;
    const int b = blockIdx.z, it = blockIdx.y, jt = blockIdx.x;
    const int i0 = it << 4, j0 = jt << 4;

    for (int idx = tid; idx < 512; idx += 256) sPar[idx] = P[idx];

    const float* __restrict__ HLb = HL + (b * NN + i0) * FF;
    const float* __restrict__ HRb = HR + (b * NN + j0) * FF;
    for (int idx = tid; idx < 16 * 32; idx += 256) {
        const int row = idx >> 5;
        const int seg = (idx & 31) << 2;
        *(float4*)&sHL[row * ROWP + seg] = *(const float4*)&HLb[row * FF + seg];
        *(float4*)&sHR[row * ROWP + seg] = *(const float4*)&HRb[row * FF + seg];
    }
    __syncthreads();

    const int ti = tid >> 4, tj = tid & 15;
    const int i = i0 + ti, j = j0 + tj;
    const size_t eidx = (((size_t)b * NN + i) * NN + j) * 2;
    const float2 e = *(const float2*)&edge[eidx];
    const float e0 = e.x, e1 = e.y;

    const float* __restrict__ hlp = &sHL[ti * ROWP];
    const float* __restrict__ hrp = &sHR[tj * ROWP];
    const float* __restrict__ we0 = sPar;
    const float* __restrict__ we1 = sPar + 128;
    const float* __restrict__ W0  = sPar + 256;
    const float* __restrict__ W1  = sPar + 384;

    float acc0 = 0.f, acc1 = 0.f;
#pragma unroll 8
    for (int f = 0; f < FF; f += 4) {
        const float4 ha = *(const float4*)(hlp + f);
        const float4 hb = *(const float4*)(hrp + f);
        const float4 wa = *(const float4*)(we0 + f);
        const float4 wb = *(const float4*)(we1 + f);
        const float4 wc = *(const float4*)(W0 + f);
        const float4 wd = *(const float4*)(W1 + f);
        float v, z;
        v = fmaf(e0, wa.x, fmaf(e1, wb.x, ha.x + hb.x));
        z = v >= 0.f ? v : NEG_SLOPE * v;
        acc0 = fmaf(wc.x, z, acc0); acc1 = fmaf(wd.x, z, acc1);
        v = fmaf(e0, wa.y, fmaf(e1, wb.y, ha.y + hb.y));
        z = v >= 0.f ? v : NEG_SLOPE * v;
        acc0 = fmaf(wc.y, z, acc0); acc1 = fmaf(wd.y, z, acc1);
        v = fmaf(e0, wa.z, fmaf(e1, wb.z, ha.z + hb.z));
        z = v >= 0.f ? v : NEG_SLOPE * v;
        acc0 = fmaf(wc.z, z, acc0); acc1 = fmaf(wd.z, z, acc1);
        v = fmaf(e0, wa.w, fmaf(e1, wb.w, ha.w + hb.w));
        z = v >= 0.f ? v : NEG_SLOPE * v;
        acc0 = fmaf(wc.w, z, acc0); acc1 = fmaf(wd.w, z, acc1);
    }

    float2 o;
    o.x = acc0 + P[512];
    o.y = acc1 + P[513];
    *(float2*)&out[eidx] = o;
}

// ---------------------------------------------------------------------------
extern "C" void kernel_launch(void* const* d_in, const int* in_sizes, int n_in,
                              void* d_out, int out_size, void* d_ws, size_t ws_size,
                              hipStream_t stream)
{
    (void)in_sizes; (void)n_in; (void)out_size; (void)ws_size;
    const float* node = (const float*)d_in[0];
    const float* edge = (const float*)d_in[1];
    const float* w1   = (const float*)d_in[2];
    const float* g1   = (const float*)d_in[3];
    const float* b1   = (const float*)d_in[4];
    const float* m1   = (const float*)d_in[5];
    const float* v1   = (const float*)d_in[6];
    const float* w2   = (const float*)d_in[7];
    const float* g2   = (const float*)d_in[8];
    const float* b2   = (const float*)d_in[9];
    const float* m2   = (const float*)d_in[10];
    const float* v2   = (const float*)d_in[11];
    const float* pw   = (const float*)d_in[12];
    const float* pb   = (const float*)d_in[13];

    float* P  = (float*)d_ws;          // 1024 floats of folded params
    float* HL = P + 1024;              // 1024*128 floats
    float* HR = HL + 1024 * FF;        // 1024*128 floats   (total ~1.03 MB)

    k_prep<<<1, 128, 0, stream>>>(w1, g1, b1, m1, v1, w2, g2, b2, m2, v2, pw, pb, P);
    k_gemm_hlr<<<256, 128, 0, stream>>>(node, w1, P, HL, HR);
    dim3 g3(NN / 16, NN / 16, 4);
    k_edge_fused<<<g3, 256, 0, stream>>>(edge, P, HL, HR, (float*)d_out);
}